// SpecAugment_78589311582796
// MI455X (gfx1250) — compile-verified
//
#include <hip/hip_runtime.h>
#include <stdint.h>

// ---------------------------------------------------------------------------
// SpecAugment for MI455X (gfx1250).
// Bandwidth-bound streaming kernel: ~262 MB total traffic -> ~11 us floor at
// 23.3 TB/s. No matrix math, so no WMMA; the CDNA5-specific path used here is
// the Tensor Data Mover (tensor_load_to_lds + s_wait_tensorcnt) to stage the
// time-mask row into LDS, plus nontemporal 128-bit global streams.
// ---------------------------------------------------------------------------

#define SA_B 64
#define SA_D 128
#define SA_T 4000
#define SA_FREQ_MASKS 2
#define SA_TIME_MASKS 10
#define SA_FREQ_WIDTH 27
#define SA_TIME_WIDTH 100

#define SA_TCHUNK 1000   // floats per t-chunk (4 chunks cover T=4000)
#define SA_DGROUP 32     // d-rows per block (4 groups cover D=128)

typedef float  f32x4 __attribute__((ext_vector_type(4)));
typedef uint32_t u32x4 __attribute__((ext_vector_type(4)));
typedef int    i32x4 __attribute__((ext_vector_type(4)));
typedef int    i32x8 __attribute__((ext_vector_type(8)));

// ----------------------------- Threefry-2x32 -------------------------------
__device__ __forceinline__ uint32_t sa_rotl(uint32_t v, int n) {
  return (v << n) | (v >> (32 - n));
}

__device__ void sa_threefry2x32(uint32_t k0, uint32_t k1, uint32_t x0, uint32_t x1,
                                uint32_t* r0, uint32_t* r1) {
  uint32_t ks2 = k0 ^ k1 ^ 0x1BD11BDAu;
  x0 += k0; x1 += k1;
#define SA_TFR(r) { x0 += x1; x1 = sa_rotl(x1, r); x1 ^= x0; }
  SA_TFR(13) SA_TFR(15) SA_TFR(26) SA_TFR(6)
  x0 += k1; x1 += ks2 + 1u;
  SA_TFR(17) SA_TFR(29) SA_TFR(16) SA_TFR(24)
  x0 += ks2; x1 += k0 + 2u;
  SA_TFR(13) SA_TFR(15) SA_TFR(26) SA_TFR(6)
  x0 += k0; x1 += k1 + 3u;
  SA_TFR(17) SA_TFR(29) SA_TFR(16) SA_TFR(24)
  x0 += k1; x1 += ks2 + 4u;
  SA_TFR(13) SA_TFR(15) SA_TFR(26) SA_TFR(6)
  x0 += ks2; x1 += k0 + 5u;
#undef SA_TFR
  *r0 = x0; *r1 = x1;
}

// random_bits(key, 32, shape) element e of n total (n even):
// counts = iota(n); pairs (i, i+n/2); out = concat(out0, out1).
__device__ uint32_t sa_random_bits(uint32_t k0, uint32_t k1, uint32_t e, uint32_t n) {
  uint32_t half = n >> 1, a, b;
  if (e < half) { sa_threefry2x32(k0, k1, e, e + half, &a, &b); return a; }
  sa_threefry2x32(k0, k1, e - half, e, &a, &b);
  return b;
}

// ----------------------- Kernel A: mask parameters -------------------------
// One thread per batch sample. Reproduces:
//   key = key(42); kf1,kf2,kt1,kt2 = split(key,4)
//   f_start = randint(kf1,(B,2),0,102); f_w = randint(kf2,(B,2),0,28)
//   u = uniform(kt1,(B,10)); t_start = floor(u*max(len-99,1)); t_w = randint(kt2,(B,10),0,101)
__global__ __launch_bounds__(64) void sa_mask_params_kernel(
    const int* __restrict__ length, float* __restrict__ fk,
    int* __restrict__ ts, int* __restrict__ te) {
  int b = threadIdx.x;
  if (b >= SA_B) return;

  // split(key(42), 4): pairs (i, i+4); key_i = (out[2i], out[2i+1])
  uint32_t o0[4], o1[4];
  #pragma unroll
  for (int i = 0; i < 4; ++i)
    sa_threefry2x32(0u, 42u, (uint32_t)i, (uint32_t)(i + 4), &o0[i], &o1[i]);
  const uint32_t kf1_0 = o0[0], kf1_1 = o0[1];
  const uint32_t kf2_0 = o0[2], kf2_1 = o0[3];
  const uint32_t kt1_0 = o1[0], kt1_1 = o1[1];
  const uint32_t kt2_0 = o1[2], kt2_1 = o1[3];

  // Frequency masks
  int fs[SA_FREQ_MASKS], fe[SA_FREQ_MASKS];
  #pragma unroll
  for (int m = 0; m < SA_FREQ_MASKS; ++m) {
    uint32_t e = (uint32_t)(SA_FREQ_MASKS * b + m);
    uint32_t bs = sa_random_bits(kf1_0, kf1_1, e, SA_B * SA_FREQ_MASKS);
    uint32_t bw = sa_random_bits(kf2_0, kf2_1, e, SA_B * SA_FREQ_MASKS);
    int start = (int)(bs % (uint32_t)(SA_D - SA_FREQ_WIDTH + 1)); // % 102
    int w     = (int)(bw % (uint32_t)(SA_FREQ_WIDTH + 1));        // % 28
    fs[m] = start; fe[m] = start + w;
  }
  for (int d = 0; d < SA_D; ++d) {
    bool hit = false;
    #pragma unroll
    for (int m = 0; m < SA_FREQ_MASKS; ++m) hit |= (d >= fs[m]) && (d < fe[m]);
    fk[b * SA_D + d] = hit ? 0.0f : 1.0f;
  }

  // Time masks
  int len = length[b];
  float max_start = (float)((len - SA_TIME_WIDTH + 1) > 1 ? (len - SA_TIME_WIDTH + 1) : 1);
  #pragma unroll
  for (int m = 0; m < SA_TIME_MASKS; ++m) {
    uint32_t e = (uint32_t)(SA_TIME_MASKS * b + m);
    uint32_t bu = sa_random_bits(kt1_0, kt1_1, e, SA_B * SA_TIME_MASKS);
    float u = __uint_as_float((bu >> 9) | 0x3f800000u) - 1.0f;  // [0,1)
    int start = (int)floorf(u * max_start);
    uint32_t bw = sa_random_bits(kt2_0, kt2_1, e, SA_B * SA_TIME_MASKS);
    int w = (int)(bw % (uint32_t)(SA_TIME_WIDTH + 1));          // % 101
    ts[b * SA_TIME_MASKS + m] = start;
    te[b * SA_TIME_MASKS + m] = start + w;
  }
}

// ---------------- Kernel B: expand time intervals -> tk[B][T] --------------
__global__ __launch_bounds__(256) void sa_time_mask_fill_kernel(
    const int* __restrict__ ts, const int* __restrict__ te, float* __restrict__ tk) {
  int idx = blockIdx.x * 256 + threadIdx.x;
  if (idx >= SA_B * SA_T) return;
  int b = idx / SA_T;
  int t = idx - b * SA_T;
  bool hit = false;
  #pragma unroll
  for (int m = 0; m < SA_TIME_MASKS; ++m)
    hit |= (t >= ts[b * SA_TIME_MASKS + m]) && (t < te[b * SA_TIME_MASKS + m]);
  tk[idx] = hit ? 0.0f : 1.0f;
}

// --------------------------- TDM 1D load to LDS ----------------------------
// D# per CDNA5 ISA ch. 8: group0 = {count/flags, lds_addr, global_addr, type=2},
// group1 carries data_size=4B, tensor_dim0 / tile_dim0 = nelem, stride0 = nelem.
__device__ __forceinline__ void sa_tdm_load_1d(uint32_t lds_off, const void* gptr,
                                               uint32_t nelem) {
#if __has_builtin(__builtin_amdgcn_tensor_load_to_lds)
  uint64_t ga = (uint64_t)(uintptr_t)gptr;
  u32x4 g0;
  g0[0] = 1u;                                               // count=1, no gather
  g0[1] = lds_off;                                          // LDS byte address
  g0[2] = (uint32_t)ga;                                     // global_addr[31:0]
  g0[3] = (uint32_t)((ga >> 32) & 0x1FFFFFFu) | (2u << 30); // addr[56:32] | type=2
  i32x8 g1;
  g1[0] = (int)(2u << 16);                                  // data_size=4B; wg_mask=0
  g1[1] = (int)((nelem & 0xFFFFu) << 16);                   // tensor_dim0[15:0]
  g1[2] = (int)((nelem >> 16) | (1u << 16));                // tensor_dim0[31:16] | tensor_dim1=1
  g1[3] = (int)((nelem & 0xFFFFu) << 16);                   // tile_dim0
  g1[4] = 0;                                                // tile_dim1=0, tile_dim2=0
  g1[5] = (int)nelem;                                       // tensor_dim0_stride low
  g1[6] = 0;
  g1[7] = 0;
  i32x4 z4 = {0, 0, 0, 0};
#if defined(__clang_major__) && (__clang_major__ >= 23)
  i32x8 z8 = {0, 0, 0, 0, 0, 0, 0, 0};
  __builtin_amdgcn_tensor_load_to_lds(g0, g1, z4, z4, z8, 0);
#else
  __builtin_amdgcn_tensor_load_to_lds(g0, g1, z4, z4, 0);
#endif
  __builtin_amdgcn_s_wait_tensorcnt(0);
#else
  (void)lds_off; (void)gptr; (void)nelem;
#endif
}

// ------------------------- Kernel C: main stream ---------------------------
// grid = (16, 64): blockIdx.x -> {chunk = x&3, dgroup = x>>2}, blockIdx.y -> b.
__global__ __launch_bounds__(256) void sa_specaug_main_kernel(
    const float* __restrict__ x, const float* __restrict__ fk,
    const float* __restrict__ tk, float* __restrict__ out) {
  __shared__ float lds_tk[SA_TCHUNK];
  __shared__ float lds_fk[SA_DGROUP];

  const int b     = blockIdx.y;
  const int chunk = blockIdx.x & 3;
  const int dg    = blockIdx.x >> 2;
  const int t0    = chunk * SA_TCHUNK;
  const int d0    = dg * SA_DGROUP;
  const int tid   = threadIdx.x;

  if (tid < SA_DGROUP) lds_fk[tid] = fk[b * SA_D + d0 + tid];

#if __has_builtin(__builtin_amdgcn_tensor_load_to_lds)
  if (tid == 0) {
    uint32_t lds_off = (uint32_t)(uintptr_t)(void*)&lds_tk[0];
    sa_tdm_load_1d(lds_off, tk + (size_t)b * SA_T + t0, SA_TCHUNK);
  }
#else
  for (int i = tid; i < SA_TCHUNK; i += 256)
    lds_tk[i] = tk[(size_t)b * SA_T + t0 + i];
#endif
  __syncthreads();

  const bool active = (tid < SA_TCHUNK / 4);   // 250 float4 per chunk
  f32x4 tkv = {0.f, 0.f, 0.f, 0.f};
  if (active) tkv = *(const f32x4*)&lds_tk[tid * 4];

  size_t base = ((size_t)(b * SA_D + d0)) * SA_T + t0;
  #pragma unroll 4
  for (int d = 0; d < SA_DGROUP; ++d) {
    float fkd = lds_fk[d];
    if (active) {
      const f32x4* px = (const f32x4*)(x + base) + tid;
      f32x4*       po = (f32x4*)(out + base) + tid;
      if (fkd != 0.0f) {
        // Row kept: stream x once (nontemporal), multiply by time keep.
        f32x4 v = __builtin_nontemporal_load(px);
        v.x *= tkv.x; v.y *= tkv.y; v.z *= tkv.z; v.w *= tkv.w;
        __builtin_nontemporal_store(v, po);
      } else {
        // Frequency-masked row: skip the read entirely, write zeros.
        f32x4 z = {0.0f, 0.0f, 0.0f, 0.0f};
        __builtin_nontemporal_store(z, po);
      }
    }
    base += SA_T;
  }
}

// ------------------------------- Launcher ----------------------------------
extern "C" void kernel_launch(void* const* d_in, const int* in_sizes, int n_in,
                              void* d_out, int out_size, void* d_ws, size_t ws_size,
                              hipStream_t stream) {
  (void)in_sizes; (void)n_in; (void)out_size; (void)ws_size;
  const float* x      = (const float*)d_in[0];   // [B, D, T] f32
  const int*   length = (const int*)d_in[1];     // [B] i32
  float*       out    = (float*)d_out;           // [B, D, T] f32

  // Workspace layout
  float* fk = (float*)d_ws;                  // [B*D]   freq keep (0/1)
  float* tk = fk + SA_B * SA_D;              // [B*T]   time keep (0/1)
  int*   ts = (int*)(tk + SA_B * SA_T);      // [B*10]  time-mask starts
  int*   te = ts + SA_B * SA_TIME_MASKS;     // [B*10]  time-mask ends

  sa_mask_params_kernel<<<1, 64, 0, stream>>>(length, fk, ts, te);

  int nBT = SA_B * SA_T;
  sa_time_mask_fill_kernel<<<(nBT + 255) / 256, 256, 0, stream>>>(ts, te, tk);

  dim3 grid(16, SA_B);   // 4 t-chunks * 4 d-groups, per batch sample
  sa_specaug_main_kernel<<<grid, 256, 0, stream>>>(x, fk, tk, out);
}